// Mean_Average_Precision_84817014161698
// MI455X (gfx1250) — compile-verified
//
#include <hip/hip_runtime.h>
#include <hip/hip_bf16.h>

// Mean Average Precision (detection) for MI455X / gfx1250, 4-kernel pipeline:
//   1) sort_kernel:   per-image stable descending bitonic sort of 2048 scores
//   2) argmax_kernel: (B x 4) grid — per-prediction first-max IoU argmax over
//                     512 GTs; GT tile DMA'd to LDS via TDM tensor_load_to_lds
//   3) match_ap_kernel: per-image greedy match (LDS ds_min atomics) + scan + AP
//   4) map_reduce_kernel: deterministic mean over images
//
// Key insight: the reference's sequential scan collapses — per GT g, the TP is
// the lowest-rank prediction whose IoU-argmax is g with IoU>0.5 and matching
// class, so matching is a per-GT min(rank) reduction.

constexpr int P  = 2048;   // predictions per image
constexpr int G  = 512;    // ground truths per image
constexpr int NT = 512;    // threads per block
constexpr int S  = 4;      // prediction chunks per image (P == S*NT)

typedef unsigned int u32x4 __attribute__((ext_vector_type(4)));
typedef int          i32x8 __attribute__((ext_vector_type(8)));
typedef int          i32x4 __attribute__((ext_vector_type(4)));

// ---------------------------------------------------------------- sort -------
__global__ __launch_bounds__(NT) void sort_kernel(
    const float* __restrict__ pred_scores,  // [B,P]
    unsigned short* __restrict__ order)     // [B,P] sorted orig indices
{
  __shared__ unsigned long long keys[P];    // 16KB
  const int b = blockIdx.x, t = threadIdx.x;
  const float* ps = pred_scores + (size_t)b * P;

  // key = score_bits<<32 | ~index  -> descending sort == stable argsort(-score)
  for (int i = t; i < P; i += NT)
    keys[i] = ((unsigned long long)__float_as_uint(ps[i]) << 32) | (unsigned)(~i);
  for (int k = 2; k <= P; k <<= 1) {
    for (int j = k >> 1; j > 0; j >>= 1) {
      __syncthreads();
      for (int i = t; i < P; i += NT) {
        int ixj = i ^ j;
        if (ixj > i) {
          unsigned long long a = keys[i], bb = keys[ixj];
          bool desc = ((i & k) == 0);
          if ((a < bb) == desc) { keys[i] = bb; keys[ixj] = a; }
        }
      }
    }
  }
  __syncthreads();
  for (int i = t; i < P; i += NT)
    order[(size_t)b * P + i] =
        (unsigned short)(~(unsigned)(keys[i] & 0xffffffffull));
}

// -------------------------------------------------------------- argmax -------
__global__ __launch_bounds__(NT) void argmax_kernel(
    const float* __restrict__ pred_boxes,    // [B,P,4] xywh
    const int*   __restrict__ pred_classes,  // [B,P]
    const float* __restrict__ gt_boxes,      // [B,G,4] xywh
    const int*   __restrict__ gt_classes,    // [B,G]
    const unsigned short* __restrict__ order,// [B,P]
    short* __restrict__ bestg_out)           // [B,P]: best GT idx or -1
{
  __shared__ float  stage[G * 4];   // 8KB TDM staging (raw xywh)
  __shared__ float4 gt4[G];         // 8KB {x1,y1,x2,y2} -> ds_load_b128
  __shared__ float4 gar[G / 4];     // 2KB areas, 4 per b128 load
  __shared__ int    gtc[G];         // 2KB class

  const int b = blockIdx.x;         // image
  const int c = blockIdx.y;         // prediction chunk
  const int t = threadIdx.x;
  const float* gbx = gt_boxes + (size_t)b * G * 4;
  const int*   gcl = gt_classes + (size_t)b * G;

#if defined(__AMDGCN__) && __has_builtin(__builtin_amdgcn_tensor_load_to_lds)
  // TDM DMA of the GT tile (G*4 = 2048 dwords, 1-D, data_size=4B) into LDS,
  // overlapped with the global order/pred-box loads below.
  if (t < 32) {  // wave 0; descriptor is uniform
    unsigned lds_off = (unsigned)(unsigned long long)(void*)stage;
    unsigned long long gaddr = (unsigned long long)(const void*)gbx;
    u32x4 g0;
    g0[0] = 1u;                                        // count=1
    g0[1] = lds_off;                                   // lds_addr (bytes)
    g0[2] = (unsigned)(gaddr & 0xffffffffu);           // global_addr[31:0]
    g0[3] = (unsigned)((gaddr >> 32) & 0x01ffffffu)    // global_addr[56:32]
          | (2u << 30);                                // type=2
    i32x8 g1;
    g1[0] = 0x00020000;           // data_size=2 (4 bytes), mask=0
    g1[1] = (int)(2048u << 16);   // tensor_dim0 = 2048
    g1[2] = 0x00010000;           // tensor_dim1 = 1
    g1[3] = (int)(2048u << 16);   // tile_dim0 = 2048
    g1[4] = 1;                    // tile_dim1 = 1
    g1[5] = 2048;                 // tensor_dim0_stride
    g1[6] = 0;  g1[7] = 0;
    i32x4 z4 = {0, 0, 0, 0};
#if __has_include(<hip/amd_detail/amd_gfx1250_TDM.h>)
    i32x8 z8 = {0, 0, 0, 0, 0, 0, 0, 0};
    __builtin_amdgcn_tensor_load_to_lds(g0, g1, z4, z4, z8, 0);
#else
    __builtin_amdgcn_tensor_load_to_lds(g0, g1, z4, z4, 0);
#endif
  }
#define MAP_USED_TDM 1
#else
  for (int i = t; i < G * 4; i += NT) stage[i] = gbx[i];
#endif

  // My prediction (global loads overlap the TDM DMA).
  const int p = c * NT + t;                       // sorted rank
  const int orig = order[(size_t)b * P + p];
  const float* bp = pred_boxes + (size_t)b * P * 4 + (size_t)orig * 4;
  __builtin_prefetch(bp, 0, 1);                   // global_prefetch_b8
  float x = bp[0], y = bp[1], w = bp[2], h = bp[3];
  float px1 = x, py1 = y, px2 = x + w, py2 = y + h;
  float pa = fabsf((px2 - px1) * (py2 - py1));
  int cls = pred_classes[(size_t)b * P + orig];

#if defined(MAP_USED_TDM)
  if (t < 32) __builtin_amdgcn_s_wait_tensorcnt(0);
#endif
  __syncthreads();

  // Convert staged xywh -> packed xyxy float4 + grouped areas + class.
  {
    float* garf = (float*)gar;
    for (int g = t; g < G; g += NT) {
      float gx = stage[4 * g + 0], gy = stage[4 * g + 1];
      float gw = stage[4 * g + 2], gh = stage[4 * g + 3];
      gt4[g]  = make_float4(gx, gy, gx + gw, gy + gh);
      garf[g] = fabsf(gw * gh);
      gtc[g]  = gcl[g];
    }
  }
  __syncthreads();

  // First-max IoU argmax over all GTs (strict > keeps first max, =jnp.argmax).
  // Blocked by 4: per 4 GTs -> 4x ds_load_b128 (boxes) + 1x ds_load_b128 (areas).
  float best = -1.0f;
  int bg = 0;
  for (int gq = 0; gq < G / 4; ++gq) {
    float4 ar = gar[gq];
    float areas[4] = {ar.x, ar.y, ar.z, ar.w};
#pragma unroll
    for (int k = 0; k < 4; ++k) {
      int g = 4 * gq + k;
      float4 q = gt4[g];
      float ix1 = fmaxf(px1, q.x);
      float iy1 = fmaxf(py1, q.y);
      float ix2 = fminf(px2, q.z);
      float iy2 = fminf(py2, q.w);
      float iw = fmaxf(ix2 - ix1, 0.0f);
      float ih = fmaxf(iy2 - iy1, 0.0f);
      float inter = iw * ih;
      float uni = pa + areas[k] - inter + 1e-9f;  // same eval order as reference
      float iou = inter / uni;                    // IEEE div: match rounding
      if (iou > best) { best = iou; bg = g; }
    }
  }
  bool cand = (best > 0.5f) && (cls == gtc[bg]);
  bestg_out[(size_t)b * P + p] = cand ? (short)bg : (short)-1;
}

// ------------------------------------------------------------- match+AP ------
__global__ __launch_bounds__(NT) void match_ap_kernel(
    const short* __restrict__ bestg_in,  // [B,P]
    float* __restrict__ aps)             // [B]
{
  __shared__ int   bg[P];     // 8KB
  __shared__ int   minp[G];   // 2KB
  __shared__ int   tp[P];     // 8KB
  __shared__ int   part[NT];  // 2KB
  __shared__ float red[NT];   // 2KB
  const int b = blockIdx.x, t = threadIdx.x;

  for (int p = t; p < P; p += NT) bg[p] = (int)bestg_in[(size_t)b * P + p];
  for (int g = t; g < G; g += NT) minp[g] = P;
  __syncthreads();

  // Per-GT lowest winning rank (LDS ds_min atomics).
  for (int p = t; p < P; p += NT) {
    int g = bg[p];
    if (g >= 0) atomicMin(&minp[g], p);
  }
  __syncthreads();
  for (int p = t; p < P; p += NT) {
    int g = bg[p];
    tp[p] = (g >= 0 && minp[g] == p) ? 1 : 0;
  }
  __syncthreads();

  // Inclusive scan of tp (4/thread + Hillis-Steele over 512 partials).
  int s0 = tp[4 * t + 0], s1 = tp[4 * t + 1];
  int s2 = tp[4 * t + 2], s3 = tp[4 * t + 3];
  part[t] = s0 + s1 + s2 + s3;
  __syncthreads();
  for (int off = 1; off < NT; off <<= 1) {
    int v = part[t];
    int u = (t >= off) ? part[t - off] : 0;
    __syncthreads();
    part[t] = v + u;
    __syncthreads();
  }
  int cum = (t == 0) ? 0 : part[t - 1];

  // AP = sum over TP preds of (prec_k + prec_{k-1}) / (2G)   (trapezoid rule).
  float contrib = 0.0f;
  int tps[4] = {s0, s1, s2, s3};
  for (int k = 0; k < 4; ++k) {
    int rank = 4 * t + k + 1;
    cum += tps[k];
    if (tps[k]) {
      float prec = (float)cum / (float)rank;
      float prev = (rank == 1) ? 1.0f : (float)(cum - 1) / (float)(rank - 1);
      contrib += 0.5f * (prec + prev);
    }
  }
  red[t] = contrib / (float)G;
  __syncthreads();
  for (int off = NT / 2; off > 0; off >>= 1) {
    if (t < off) red[t] += red[t + off];
    __syncthreads();
  }
  if (t == 0) aps[b] = red[0];
}

// --------------------------------------------------------------- reduce ------
__global__ void map_reduce_kernel(const float* __restrict__ aps,
                                  float* __restrict__ out, int B) {
  __shared__ float s[256];
  int t = threadIdx.x;
  float v = 0.0f;
  for (int i = t; i < B; i += 256) v += aps[i];
  s[t] = v;
  __syncthreads();
  for (int off = 128; off > 0; off >>= 1) {
    if (t < off) s[t] += s[t + off];
    __syncthreads();
  }
  if (t == 0) out[0] = s[0] / (float)B;
}

// --------------------------------------------------------------- launch ------
extern "C" void kernel_launch(void* const* d_in, const int* in_sizes, int n_in,
                              void* d_out, int out_size, void* d_ws, size_t ws_size,
                              hipStream_t stream) {
  const float* pred_boxes   = (const float*)d_in[0];
  const float* pred_scores  = (const float*)d_in[1];
  const int*   pred_classes = (const int*)d_in[2];
  const float* gt_boxes     = (const float*)d_in[3];
  const int*   gt_classes   = (const int*)d_in[4];

  const int B = in_sizes[1] / P;  // pred_scores is [B,P]

  // Workspace layout: order[B*P] u16 | bestg[B*P] i16 | aps[B] f32
  unsigned short* order = (unsigned short*)d_ws;
  short* bestg = (short*)(order + (size_t)B * P);
  float* aps   = (float*)(bestg + (size_t)B * P);

  sort_kernel<<<B, NT, 0, stream>>>(pred_scores, order);
  argmax_kernel<<<dim3(B, S), NT, 0, stream>>>(pred_boxes, pred_classes,
                                               gt_boxes, gt_classes, order,
                                               bestg);
  match_ap_kernel<<<B, NT, 0, stream>>>(bestg, aps);
  map_reduce_kernel<<<1, 256, 0, stream>>>(aps, (float*)d_out, B);
}